// Cabasc_24421184045639
// MI455X (gfx1250) — compile-verified
//
#include <hip/hip_runtime.h>
#include <hip/hip_bf16.h>

// ---------------------------------------------------------------------------
// Cabasc forward for MI455X (gfx1250), fp32 WMMA path.
//   B=128, S=256, A=8, E=H=300, V=50000, P=3
// Matrix ops use V_WMMA_F32_16X16X4_F32 (exact fp32). Recurrent GRU runs as a
// single persistent kernel: 16 workgroups (2 GRUs x 8 batch tiles of 16),
// each iterating all 256 timesteps with workgroup barriers only.
// Round-2 change: out-of-range output columns are handled by CLAMPING the
// weight-row index (loads always in-bounds) instead of predicated loads, so
// the hot K-loop has no exec-mask save/restore and no zero-fill moves; the
// six B pointers are hoisted out of the K-loop for base+imm addressing.
// ---------------------------------------------------------------------------

typedef float v2f __attribute__((ext_vector_type(2)));
typedef float v8f __attribute__((ext_vector_type(8)));

constexpr int kB = 128;
constexpr int kS = 256;
constexpr int kA = 8;
constexpr int kE = 300;
constexpr int kH = 300;
constexpr int kP = 3;
constexpr int kK = 900;     // 3E = 3H
constexpr int kAP = 304;    // padded tile width (16-mult, even)
constexpr int kKP = 904;    // padded feat K width (even)

__device__ __forceinline__ int imin(int a, int b) { return a < b ? a : b; }
__device__ __forceinline__ int imax(int a, int b) { return a > b ? a : b; }
__device__ __forceinline__ float sigmoidf(float x) { return 1.f / (1.f + __expf(-x)); }

// ---------------------------------------------------------------------------
// K1: lengths (memory_len, aspect_len, left_len) and v_a (aspect avg embed)
// ---------------------------------------------------------------------------
__global__ __launch_bounds__(320)
void prep_kernel(const float* __restrict__ embed, const int* __restrict__ text,
                 const int* __restrict__ aspect, const int* __restrict__ xl,
                 int* __restrict__ lens, float* __restrict__ v_a)
{
    const int b = blockIdx.x;
    const int tid = threadIdx.x;
    __shared__ int cnt[3];
    if (tid < 3) cnt[tid] = 0;
    __syncthreads();
    if (tid < kS) {
        if (text[b * kS + tid] != 0) atomicAdd(&cnt[0], 1);
        if (xl[b * kS + tid] != 0)   atomicAdd(&cnt[2], 1);
    }
    if (tid < kA) {
        if (aspect[b * kA + tid] != 0) atomicAdd(&cnt[1], 1);
    }
    __syncthreads();
    if (tid < 3) lens[tid * kB + b] = cnt[tid];
    const float inv = 1.f / (float)imax(cnt[1], 1);
    for (int e = tid; e < kE; e += blockDim.x) {
        float s = 0.f;
        for (int a = 0; a < kA; ++a)
            s += embed[(long)aspect[b * kA + a] * kE + e];
        v_a[b * kE + e] = s * inv;
    }
}

// ---------------------------------------------------------------------------
// K2: both GRUs + per-step attention scalar.  grid = 16 blocks:
//     blockIdx>>3 selects side (0=left,1=right), blockIdx&7 selects 16-row
//     batch tile.  512 threads = 16 waves.  Each step:
//       gather emb tile -> WMMA f32 GEMMs (gi, gh fused per gate) ->
//       fused gate math in registers -> h double-buffer swap -> attn dot.
// ---------------------------------------------------------------------------
__global__ __launch_bounds__(512)
void gru_attn_kernel(const float* __restrict__ embed,
                     const int* __restrict__ xl, const int* __restrict__ xr,
                     const float* __restrict__ wih_l, const float* __restrict__ whh_l,
                     const float* __restrict__ bih_l, const float* __restrict__ bhh_l,
                     const float* __restrict__ wih_r, const float* __restrict__ whh_r,
                     const float* __restrict__ bih_r, const float* __restrict__ bhh_r,
                     const float* __restrict__ wml, const float* __restrict__ bml,
                     const float* __restrict__ wmr, const float* __restrict__ bmr,
                     float* __restrict__ attn_l, float* __restrict__ attn_r)
{
    __shared__ __align__(16) float hbuf[2][16][kAP];   // h double buffer
    __shared__ __align__(16) float eT[16][kAP];        // embedding tile

    const int side = blockIdx.x >> 3;
    const int b0   = (blockIdx.x & 7) << 4;
    const int*   xs  = side ? xr    : xl;
    const float* wih = side ? wih_r : wih_l;
    const float* whh = side ? whh_r : whh_l;
    const float* bih = side ? bih_r : bih_l;
    const float* bhh = side ? bhh_r : bhh_l;
    const float* wmv = side ? wmr   : wml;
    const float  bm0 = side ? bmr[0] : bml[0];
    float* attn      = side ? attn_r : attn_l;

    const int tid  = threadIdx.x;
    const int lane = tid & 31;
    const int wv   = tid >> 5;            // 0..15
    const int rowA = lane & 15;           // M-row / N-col within fragment
    const int kk   = (lane >> 4) << 1;    // K sub-offset (0 or 2)
    const int mb   = (lane >> 4) << 3;    // C fragment M base (0 or 8)

    for (int i = tid; i < 2 * 16 * kAP; i += 512) (&hbuf[0][0][0])[i] = 0.f;
    __syncthreads();

    int cur = 0;
    for (int t = 0; t < kS; ++t) {
        // gather embedding rows for the 16 batch lanes at timestep t
        for (int i = tid; i < 16 * kE; i += 512) {
            int r = i / kE, e = i - r * kE;
            eT[r][e] = embed[(long)xs[(b0 + r) * kS + t] * kE + e];
        }
        __syncthreads();

        const float (*hc)[kAP] = hbuf[cur];
        float (*hn_)[kAP]      = hbuf[cur ^ 1];

        for (int jt = wv; jt < 19; jt += 16) {           // 19 column tiles (H=300)
            const int  j0  = jt << 4;
            const int  jc  = j0 + rowA;                  // output column
            const bool jok = jc < kH;
            // clamp: out-of-range lanes compute a harmless duplicate column
            // that is never stored -> no divergence in the hot loop.
            const int  jcl = imin(jc, kH - 1);
            const float* pri = wih + (long)jcl * kE + kk;
            const float* prh = whh + (long)jcl * kE + kk;
            const float* pzi = wih + (long)(kH + jcl) * kE + kk;
            const float* pzh = whh + (long)(kH + jcl) * kE + kk;
            const float* pni = wih + (long)(2 * kH + jcl) * kE + kk;
            const float* pnh = whh + (long)(2 * kH + jcl) * kE + kk;
            v8f cr = {}, cz = {}, cin = {}, chn = {};
            for (int k0 = 0; k0 < kE; k0 += 4) {
                const int ka = k0 + kk;
                v2f ae, ah;
                ae[0] = eT[rowA][ka]; ae[1] = eT[rowA][ka + 1];
                ah[0] = hc[rowA][ka]; ah[1] = hc[rowA][ka + 1];
                v2f bri, brh, bzi, bzh, bni, bnh;
                float2 v;
                v = *(const float2*)(pri + k0); bri[0] = v.x; bri[1] = v.y;
                v = *(const float2*)(prh + k0); brh[0] = v.x; brh[1] = v.y;
                v = *(const float2*)(pzi + k0); bzi[0] = v.x; bzi[1] = v.y;
                v = *(const float2*)(pzh + k0); bzh[0] = v.x; bzh[1] = v.y;
                v = *(const float2*)(pni + k0); bni[0] = v.x; bni[1] = v.y;
                v = *(const float2*)(pnh + k0); bnh[0] = v.x; bnh[1] = v.y;
                // r,z gates: gi+gh fused into one accumulator; n gate split.
                cr  = __builtin_amdgcn_wmma_f32_16x16x4_f32(false, ae, false, bri, (short)0, cr,  false, false);
                cr  = __builtin_amdgcn_wmma_f32_16x16x4_f32(false, ah, false, brh, (short)0, cr,  false, false);
                cz  = __builtin_amdgcn_wmma_f32_16x16x4_f32(false, ae, false, bzi, (short)0, cz,  false, false);
                cz  = __builtin_amdgcn_wmma_f32_16x16x4_f32(false, ah, false, bzh, (short)0, cz,  false, false);
                cin = __builtin_amdgcn_wmma_f32_16x16x4_f32(false, ae, false, bni, (short)0, cin, false, false);
                chn = __builtin_amdgcn_wmma_f32_16x16x4_f32(false, ah, false, bnh, (short)0, chn, false, false);
            }
            if (jok) {
                const float b_r  = bih[jc] + bhh[jc];
                const float b_z  = bih[kH + jc] + bhh[kH + jc];
                const float b_ni = bih[2 * kH + jc];
                const float b_nh = bhh[2 * kH + jc];
                for (int i = 0; i < 8; ++i) {
                    const int m = mb + i;
                    float r = sigmoidf(cr[i] + b_r);
                    float z = sigmoidf(cz[i] + b_z);
                    float n = tanhf(cin[i] + b_ni + r * (chn[i] + b_nh));
                    hn_[m][jc] = (1.f - z) * n + z * hc[m][jc];
                }
            }
        }
        __syncthreads();

        // attn[b,t] = sigmoid(dot(h_t, wm) + bm) + 0.5 : wave wv reduces row wv
        float p = 0.f;
        for (int j = lane; j < kH; j += 32) p += hn_[wv][j] * wmv[j];
        for (int off = 16; off; off >>= 1) p += __shfl_xor(p, off, 32);
        if (lane == 0) attn[(b0 + wv) * kS + t] = sigmoidf(p + bm0) + 0.5f;
        __syncthreads();
        cur ^= 1;
    }
}

// ---------------------------------------------------------------------------
// K3: position weights w[b,s] and v_s = sum_s(embed*w)/memory_len
// ---------------------------------------------------------------------------
__global__ __launch_bounds__(256)
void wvs_kernel(const float* __restrict__ embed, const int* __restrict__ text,
                const int* __restrict__ lens,
                const float* __restrict__ attn_l, const float* __restrict__ attn_r,
                float* __restrict__ wbuf, float* __restrict__ v_s)
{
    const int b = blockIdx.x, tid = threadIdx.x;
    __shared__ float wsh[kS];
    __shared__ int idxs[kS];
    const int memlen  = lens[0 * kB + b];
    const int asplen  = lens[1 * kB + b];
    const int leftlen = lens[2 * kB + b];
    const int a_start = leftlen - asplen;
    const int a_end   = leftlen;
    {
        const int i = tid;
        const int i_rr = imin(imax(a_end - 1 - i, 0), kS - 1);
        const int i_ls = imin(imax(i - a_start, 0), kS - 1);
        const float rr = attn_r[b * kS + i_rr];
        const float ls = attn_l[b * kS + i_ls];
        float w = (i < a_start) ? rr : ((i < a_end) ? 0.5f * (rr + ls) : ls);
        if (i >= memlen) w = 1.f;
        wsh[i] = w;
        wbuf[b * kS + i] = w;
        idxs[i] = text[b * kS + i];
    }
    __syncthreads();
    const float inv = 1.f / (float)imax(memlen, 1);
    for (int e = tid; e < kE; e += 256) {
        float s = 0.f;
        for (int sI = 0; sI < kS; ++sI)
            s += embed[(long)idxs[sI] * kE + e] * wsh[sI];
        v_s[b * kE + e] = s * inv;
    }
}

// ---------------------------------------------------------------------------
// K4: c[b,s] = tanh(feat @ w1^T + b1) @ w2^T, feat = [mem_w | v_a | v_s].
// One block per 16-row M-tile; feat tile built on the fly in LDS; WMMA f32
// over K=900, fused tanh*w2 reduction into per-row scalars.
// ---------------------------------------------------------------------------
__global__ __launch_bounds__(512)
void feat_kernel(const float* __restrict__ embed, const int* __restrict__ text,
                 const float* __restrict__ wbuf, const float* __restrict__ v_a,
                 const float* __restrict__ v_s,
                 const float* __restrict__ w1, const float* __restrict__ b1,
                 const float* __restrict__ w2, float* __restrict__ cbuf)
{
    __shared__ __align__(16) float AT[16][kKP];
    __shared__ float csh[16];

    const int mt   = blockIdx.x;           // 0..2047
    const int tid  = threadIdx.x;
    const int lane = tid & 31;
    const int wv   = tid >> 5;
    const int rowA = lane & 15;
    const int kk   = (lane >> 4) << 1;
    const int mb   = (lane >> 4) << 3;

    for (int i = tid; i < 16 * kK; i += 512) {
        const int r = i / kK, k = i - r * kK;
        const int m = (mt << 4) + r;
        const int b = m >> 8, s = m & 255;
        float v;
        if (k < kE)          v = embed[(long)text[b * kS + s] * kE + k] * wbuf[b * kS + s];
        else if (k < 2 * kE) v = v_a[b * kE + (k - kE)];
        else                 v = v_s[b * kE + (k - 2 * kE)];
        AT[r][k] = v;
    }
    if (tid < 16) csh[tid] = 0.f;
    __syncthreads();

    for (int qt = wv; qt < 19; qt += 16) {          // 19 tiles cover N=300
        const int  q0  = qt << 4;
        const int  qc  = q0 + rowA;
        const bool qok = qc < kE;
        const int  qcl = imin(qc, kE - 1);          // clamped, always in-bounds
        const float* pw = w1 + (long)qcl * kK + kk;
        v8f acc = {};
        for (int k0 = 0; k0 < kK; k0 += 4) {
            const int ka = k0 + kk;
            v2f a, bfr;
            a[0] = AT[rowA][ka]; a[1] = AT[rowA][ka + 1];
            float2 v = *(const float2*)(pw + k0);
            bfr[0] = v.x; bfr[1] = v.y;
            acc = __builtin_amdgcn_wmma_f32_16x16x4_f32(false, a, false, bfr, (short)0, acc, false, false);
        }
        const float w2q = qok ? w2[qc] : 0.f;       // kills duplicate columns
        const float b1q = qok ? b1[qc] : 0.f;
        for (int i = 0; i < 8; ++i) {
            const float hv = tanhf(acc[i] + b1q) * w2q;
            atomicAdd(&csh[mb + i], hv);
        }
    }
    __syncthreads();
    if (tid < 16) cbuf[(mt << 4) + tid] = csh[tid];
}

// ---------------------------------------------------------------------------
// K5: softmax over c -> alpha; v_ns = sum_s(embed*w*alpha) + v_s
// ---------------------------------------------------------------------------
__global__ __launch_bounds__(256)
void alpha_vns_kernel(const float* __restrict__ embed, const int* __restrict__ text,
                      const float* __restrict__ cbuf, const float* __restrict__ wbuf,
                      const float* __restrict__ v_s, float* __restrict__ v_ns)
{
    const int b = blockIdx.x, tid = threadIdx.x;
    __shared__ float fsh[kS];
    __shared__ int idxs[kS];
    __shared__ float stat[2];
    fsh[tid]  = cbuf[b * kS + tid];
    idxs[tid] = text[b * kS + tid];
    __syncthreads();
    if (tid == 0) {
        float mx = -1e30f;
        for (int s = 0; s < kS; ++s) mx = fmaxf(mx, fsh[s]);
        float sum = 0.f;
        for (int s = 0; s < kS; ++s) sum += __expf(fsh[s] - mx);
        stat[0] = mx; stat[1] = 1.f / sum;
    }
    __syncthreads();
    const float f = __expf(fsh[tid] - stat[0]) * stat[1] * wbuf[b * kS + tid];
    __syncthreads();
    fsh[tid] = f;
    __syncthreads();
    for (int e = tid; e < kE; e += 256) {
        float s = 0.f;
        for (int sI = 0; sI < kS; ++sI)
            s += embed[(long)idxs[sI] * kE + e] * fsh[sI];
        v_ns[b * kE + e] = s + v_s[b * kE + e];
    }
}

// ---------------------------------------------------------------------------
// K6: v_ms = tanh(v_ns @ wm^T + bm); out = softmax(v_ms @ wd^T + bd)
// ---------------------------------------------------------------------------
__global__ __launch_bounds__(320)
void final_kernel(const float* __restrict__ v_ns, const float* __restrict__ wm,
                  const float* __restrict__ bm, const float* __restrict__ wd,
                  const float* __restrict__ bd, float* __restrict__ out)
{
    const int b = blockIdx.x, tid = threadIdx.x;
    __shared__ float vn[kE];
    __shared__ float vm[kE];
    __shared__ float lg[kP];
    for (int e = tid; e < kE; e += 320) vn[e] = v_ns[b * kE + e];
    __syncthreads();
    for (int q = tid; q < kE; q += 320) {
        float s = bm[q];
        for (int e = 0; e < kE; ++e) s += vn[e] * wm[(long)q * kE + e];
        vm[q] = tanhf(s);
    }
    __syncthreads();
    if (tid < kP) {
        float s = bd[tid];
        for (int q = 0; q < kE; ++q) s += vm[q] * wd[(long)tid * kE + q];
        lg[tid] = s;
    }
    __syncthreads();
    if (tid == 0) {
        const float mx = fmaxf(lg[0], fmaxf(lg[1], lg[2]));
        const float e0 = __expf(lg[0] - mx), e1 = __expf(lg[1] - mx), e2 = __expf(lg[2] - mx);
        const float inv = 1.f / (e0 + e1 + e2);
        out[b * kP + 0] = e0 * inv;
        out[b * kP + 1] = e1 * inv;
        out[b * kP + 2] = e2 * inv;
    }
}

// ---------------------------------------------------------------------------
extern "C" void kernel_launch(void* const* d_in, const int* in_sizes, int n_in,
                              void* d_out, int out_size, void* d_ws, size_t ws_size,
                              hipStream_t stream)
{
    const float* embed  = (const float*)d_in[0];
    const float* w1     = (const float*)d_in[1];
    const float* b1     = (const float*)d_in[2];
    const float* w2     = (const float*)d_in[3];
    const float* wm     = (const float*)d_in[4];
    const float* bm     = (const float*)d_in[5];
    const float* wd     = (const float*)d_in[6];
    const float* bd     = (const float*)d_in[7];
    const float* wih_l  = (const float*)d_in[8];
    const float* whh_l  = (const float*)d_in[9];
    const float* bih_l  = (const float*)d_in[10];
    const float* bhh_l  = (const float*)d_in[11];
    const float* wih_r  = (const float*)d_in[12];
    const float* whh_r  = (const float*)d_in[13];
    const float* bih_r  = (const float*)d_in[14];
    const float* bhh_r  = (const float*)d_in[15];
    const float* wml    = (const float*)d_in[16];
    const float* bml    = (const float*)d_in[17];
    const float* wmr    = (const float*)d_in[18];
    const float* bmr    = (const float*)d_in[19];
    const int*   text   = (const int*)d_in[20];
    const int*   aspect = (const int*)d_in[21];
    const int*   xl     = (const int*)d_in[22];
    const int*   xr     = (const int*)d_in[23];
    float* out = (float*)d_out;

    float* ws    = (float*)d_ws;
    float* f_va  = ws;  ws += kB * kE;
    float* f_atl = ws;  ws += kB * kS;
    float* f_atr = ws;  ws += kB * kS;
    float* f_w   = ws;  ws += kB * kS;
    float* f_vs  = ws;  ws += kB * kE;
    float* f_c   = ws;  ws += kB * kS;
    float* f_vns = ws;  ws += kB * kE;
    int*   i_len = (int*)ws;

    prep_kernel<<<kB, 320, 0, stream>>>(embed, text, aspect, xl, i_len, f_va);
    gru_attn_kernel<<<16, 512, 0, stream>>>(embed, xl, xr,
        wih_l, whh_l, bih_l, bhh_l, wih_r, whh_r, bih_r, bhh_r,
        wml, bml, wmr, bmr, f_atl, f_atr);
    wvs_kernel<<<kB, 256, 0, stream>>>(embed, text, i_len, f_atl, f_atr, f_w, f_vs);
    feat_kernel<<<(kB * kS) / 16, 512, 0, stream>>>(embed, text, f_w, f_va, f_vs,
                                                    w1, b1, w2, f_c);
    alpha_vns_kernel<<<kB, 256, 0, stream>>>(embed, text, f_c, f_w, f_vs, f_vns);
    final_kernel<<<kB, 320, 0, stream>>>(f_vns, wm, bm, wd, bd, out);
}